// PhaseAdaptiveInput_12970801234492
// MI455X (gfx1250) — compile-verified
//
#include <hip/hip_runtime.h>

typedef __attribute__((ext_vector_type(2))) float v2f;
typedef __attribute__((ext_vector_type(8))) float v8f;

#define LPA 128
#define COUNT 6
#define BUCKET_SIZE 5
#define K_ACT 32
#define WAVES_PER_BLOCK 8

// One wave32 per sample. out(1x128) = values(1x32) x gatheredW(32x128), done as
// 8 column tiles x 8 K-chunks of V_WMMA_F32_16X16X4_F32, then bias + clamp^2.
__global__ __launch_bounds__(256) void PhaseAdaptiveInput_kernel(
    const int* __restrict__ fidx,      // [B, 32]
    const float* __restrict__ vals,    // [B, 32]
    const int* __restrict__ ply,       // [B]
    const float* __restrict__ weight,  // [F, OUT] row-major, OUT = 768
    const float* __restrict__ bias,    // [OUT]
    float* __restrict__ out,           // [B, 128]
    int B, int outStride)
{
  const int lane   = threadIdx.x & 31;
  const int wave   = threadIdx.x >> 5;
  const int sample = blockIdx.x * WAVES_PER_BLOCK + wave;
  if (sample >= B) return;   // wave-uniform guard; EXEC stays all-1s inside

  // Bucket selection commutes with the linear op: only compute 128 columns.
  int p = ply[sample];
  int bucket = p / BUCKET_SIZE;
  if (bucket > COUNT - 1) bucket = COUNT - 1;
  const int base = bucket * LPA;

  // Lane l owns value/index of active feature l.
  const float v  = vals[(long)sample * K_ACT + lane];
  const int   fi = fidx[(long)sample * K_ACT + lane];

  // Precompute per-K-chunk A fragments and gather rows.
  // Chunk k covers features 4k..4k+3. A(16x4 f32) layout: lanes 0-15 hold
  // K=0/1 in VGPR0/1, lanes 16-31 hold K=2/3. Only M=0 (lanes 0 and 16) live.
  const int h2 = (lane >> 4) * 2;                  // 0 for K0/K1 half, 2 for K2/K3 half
  const bool a_live = (lane & 15) == 0;            // lanes 0 and 16 carry row M=0
  float ax[8], ay[8];
  int   rx[8], ry[8];
  #pragma unroll
  for (int k = 0; k < 8; ++k) {
    const int s = 4 * k + h2;
    const float a0 = __shfl(v, s,     32);
    const float a1 = __shfl(v, s + 1, 32);
    rx[k] = __shfl(fi, s,     32);
    ry[k] = __shfl(fi, s + 1, 32);
    ax[k] = a_live ? a0 : 0.0f;
    ay[k] = a_live ? a1 : 0.0f;
  }

  const int colInTile = lane & 15;
  float res[8];

  #pragma unroll
  for (int t = 0; t < 8; ++t) {
    const int col = base + t * 16 + colInTile;
    v8f acc = {0.f, 0.f, 0.f, 0.f, 0.f, 0.f, 0.f, 0.f};
    #pragma unroll
    for (int k = 0; k < 8; ++k) {
      v2f a; a[0] = ax[k]; a[1] = ay[k];
      // B(4x16 f32): lanes 0-15 hold rows K=0/1 (VGPR0/1), lanes 16-31 K=2/3.
      // Each half-wave load is a contiguous 64B segment of a weight row.
      v2f b;
      b[0] = weight[(long)rx[k] * outStride + col];
      b[1] = weight[(long)ry[k] * outStride + col];
      acc = __builtin_amdgcn_wmma_f32_16x16x4_f32(
          /*neg_a=*/false, a, /*neg_b=*/false, b,
          /*c_mod=*/(short)0, acc, /*reuse_a=*/false, /*reuse_b=*/false);
    }
    res[t] = acc[0];  // D row M=0 = VGPR0, lanes 0..15
  }

  // Epilogue (divergent, all WMMAs already done): bias + clamp-square.
  if (lane < 16) {
    #pragma unroll
    for (int t = 0; t < 8; ++t) {
      const int c = t * 16 + lane;
      float x = res[t] + bias[base + c];
      x = fminf(fmaxf(x, 0.0f), 1.0f);
      x = x * x * (1023.0f / 1024.0f);
      out[(long)sample * LPA + c] = x;
    }
  }
}

extern "C" void kernel_launch(void* const* d_in, const int* in_sizes, int n_in,
                              void* d_out, int out_size, void* d_ws, size_t ws_size,
                              hipStream_t stream) {
  // setup_inputs() order:
  // 0: feature_indices [B,32] i32, 1: values [B,32] f32, 2: batch_size (dev scalar),
  // 3: in_features (dev scalar), 4: ply [B] i32, 5: weight [F,768] f32, 6: bias [768] f32
  const int*   fidx   = (const int*)d_in[0];
  const float* vals   = (const float*)d_in[1];
  const int*   ply    = (const int*)d_in[4];
  const float* weight = (const float*)d_in[5];
  const float* bias   = (const float*)d_in[6];
  float* out = (float*)d_out;

  const int B   = in_sizes[4];   // ply has one entry per sample
  const int OUT = in_sizes[6];   // bias length = LPA*COUNT = 768 (weight row stride)

  const int blocks = (B + WAVES_PER_BLOCK - 1) / WAVES_PER_BLOCK;
  PhaseAdaptiveInput_kernel<<<blocks, WAVES_PER_BLOCK * 32, 0, stream>>>(
      fidx, vals, ply, weight, bias, out, B, OUT);
}